// Electro_net_33741263078052
// MI455X (gfx1250) — compile-verified
//
#include <hip/hip_runtime.h>
#include <hip/hip_bf16.h>
#include <math.h>

#define ELECTRO_TYPES 200

typedef int v4i __attribute__((ext_vector_type(4)));

// CDNA5 async global->LDS path (ASYNCcnt), guarded so the source still
// compiles if this toolchain spells the builtins differently.
#if defined(__gfx1250__) && __has_builtin(__builtin_amdgcn_global_load_async_to_lds_b128) && __has_builtin(__builtin_amdgcn_s_wait_asynccnt)
#define USE_ASYNC_LDS 1
#else
#define USE_ASYNC_LDS 0
#endif

__global__ __launch_bounds__(256) void electro_pairs_kernel(
    const float* __restrict__ coords,      // [N_ATOMS,3]
    const float* __restrict__ partners,    // [N_ATOMS,2,3]
    const float* __restrict__ props,       // [N_TYPES,5]
    const int*   __restrict__ desc,        // [N_ATOMS,3]
    const int*   __restrict__ pairs,       // [N_PAIRS,2]
    float*       __restrict__ out,         // [N_PAIRS]
    int n_pairs)
{
  // ---- stage the 200x5 property table into LDS (4 KB) ----
  __shared__ __align__(16) float shProps[ELECTRO_TYPES * 5];
  {
    const int nvec = (ELECTRO_TYPES * 5) / 4;   // 250 x float4
    const int t = threadIdx.x;
    if (t < nvec) {
#if USE_ASYNC_LDS
      __builtin_amdgcn_global_load_async_to_lds_b128(
          (__attribute__((address_space(1))) v4i*)(props + 4 * t),
          (__attribute__((address_space(3))) v4i*)(shProps + 4 * t),
          0, 0);
#else
      float4 v = ((const float4*)props)[t];
      ((float4*)shProps)[t] = v;
#endif
    }
#if USE_ASYNC_LDS
    __builtin_amdgcn_s_wait_asynccnt(0);
#endif
    __syncthreads();
  }

  const int p = (int)(blockIdx.x * blockDim.x + threadIdx.x);
  if (p >= n_pairs) return;

  // Prefetch the streamed pair-index array well ahead (global_prefetch_b8;
  // speculative, OOB addresses are silently dropped).
  __builtin_prefetch(pairs + 2 * (size_t)p + 32768, 0, 0);

  const int2 pr = ((const int2*)pairs)[p];
  const int i = pr.x;
  const int j = pr.y;

  // ---- descriptor + coordinate gathers (L2 resident) ----
  const int at1 = desc[3 * i + 0];
  const int rn1 = desc[3 * i + 1];
  const int ch1 = desc[3 * i + 2];
  const int at2 = desc[3 * j + 0];
  const int rn2 = desc[3 * j + 1];
  const int ch2 = desc[3 * j + 2];

  const float cix = coords[3 * i + 0], ciy = coords[3 * i + 1], ciz = coords[3 * i + 2];
  const float cjx = coords[3 * j + 0], cjy = coords[3 * j + 1], cjz = coords[3 * j + 2];

  const float dx = cix - cjx, dy = ciy - cjy, dz = ciz - cjz;
  const float dist = sqrtf(dx * dx + dy * dy + dz * dz + 1e-12f);

  // ---- masks ----
  const bool rc1 = (at1 == 10) || (at1 == 11);
  const bool rc2 = (at2 == 10) || (at2 == 11);
  const bool argInv = ((at1 == 12) && ((at2 == 12) || rc2)) ||
                      ((at2 == 12) && ((at1 == 12) || rc1));

  const float* P1 = shProps + at1 * 5;   // LDS lookups
  const float* P2 = shProps + at2 * 5;
  const float c1 = P1[1];
  const float c2 = P2[1];

  const bool is_charged = ((P1[0] == 1.0f) && (P2[0] == 1.0f)) || argInv;
  const bool diff_res   = (rn1 != rn2) || (ch1 != ch2);
  const bool valid      = is_charged && (dist <= 15.0f) && diff_res;

  const float CONSTANT_ = expf(-0.004314f * (298.0f - 273.0f)); // folded

  float energy = 0.0f;

  // ---- kon term: charged-charged, non-dipole, inter-chain (~5% of lanes) ----
  {
    const bool cc = valid && (c1 != 0.0f) && (c2 != 0.0f) &&
                    (c1 != 999.0f) && (c2 != 999.0f) &&
                    (P1[2] != 1.0f) && (P2[2] != 1.0f);
    const bool kon_mask = cc && (P1[3] == 0.0f) && (P2[3] == 0.0f) && (ch1 != ch2);
    if (kon_mask) {
      const float sum_radii = P1[4] + P2[4] - 0.09f;
      const float dd = fmaxf(dist, sum_radii);
      const float temp_kon = fmaxf(dd + dd * dd / 30.0f, 6.0f);
      const float k_on = sqrtf(200.0f * fabsf(c1) * fabsf(c2) * 0.05f / 298.0f);
      const float firstNumber  = 332.0f * c1 * c2 / (88.0f * temp_kon * CONSTANT_);
      const float secondNumber = expf(-k_on * (temp_kon - 6.0f));
      const float thirdSecond  = 1.0f + k_on * 6.0f;
      energy += 0.5f * firstNumber * secondNumber / thirdSecond;
    }
  }

  // ---- Arg geometric term (~0.015% of lanes; exec-mask branch skips the
  //      partner gathers + acosf for almost every wave) ----
  if (valid && argInv) {
    const float u1x = partners[6 * i + 0] - cix;
    const float u1y = partners[6 * i + 1] - ciy;
    const float u1z = partners[6 * i + 2] - ciz;
    const float w1x = partners[6 * i + 3] - cix;
    const float w1y = partners[6 * i + 4] - ciy;
    const float w1z = partners[6 * i + 5] - ciz;
    const float n1x = u1y * w1z - u1z * w1y;
    const float n1y = u1z * w1x - u1x * w1z;
    const float n1z = u1x * w1y - u1y * w1x;

    const float u2x = partners[6 * j + 0] - cjx;
    const float u2y = partners[6 * j + 1] - cjy;
    const float u2z = partners[6 * j + 2] - cjz;
    const float w2x = partners[6 * j + 3] - cjx;
    const float w2y = partners[6 * j + 4] - cjy;
    const float w2z = partners[6 * j + 5] - cjz;
    const float n2x = u2y * w2z - u2z * w2y;
    const float n2y = u2z * w2x - u2x * w2z;
    const float n2z = u2x * w2y - u2y * w2x;

    const float nn1 = sqrtf(n1x * n1x + n1y * n1y + n1z * n1z + 1e-12f);
    const float nn2 = sqrtf(n2x * n2x + n2y * n2y + n2z * n2z + 1e-12f);
    const bool  m   = (nn1 > 1e-6f) && (nn2 > 1e-6f);

    float cosang = (n1x * n2x + n1y * n2y + n1z * n2z) / (nn1 * nn2);
    cosang = fminf(fmaxf(cosang, -0.999999f), 0.999999f);
    const float ang = acosf(cosang);

    const float RAD10_  = 0.17453292519943295f;
    const float RAD30_  = 0.5235987755982988f;
    const float RAD150_ = 2.6179938779914944f;
    const float RAD180_ = 3.14159265358979323f;

    const bool ang_bad = ((ang > RAD30_) && (ang < RAD150_)) ||
                         (dist < 3.9f) || (dist > 4.3f) || !m;
    if (!ang_bad) {
      const float dsafe   = fmaxf(dist, 1e-6f);
      const float argEpss = 332.0f / (dsafe * 8.8f * CONSTANT_);
      const float temp_d  = 3.0f + fabsf(dist - 4.1f);
      const bool  angCorr = dist > RAD150_;   // faithful to reference (dist vs radians)
      const float corr_ang = angCorr ? fmaxf((RAD180_ - ang) / RAD10_, 1.0f)
                                     : fmaxf(dist / RAD10_, 1.0f);
      const bool  arg_arg = (at1 == 13) && (at2 == 13);
      const float c2e = arg_arg ? -1.0f : -0.5f;
      const float ION_CORR_ = 0.02f + 0.05f / 1.4f;
      const float vk = sqrtf(200.0f * fabsf(c2e) * ION_CORR_ / 298.0f);
      energy += (1.0f / corr_ang) * c2e * argEpss *
                expf(-temp_d * vk) / (temp_d * temp_d);
    }
  }

  out[p] = energy;
}

extern "C" void kernel_launch(void* const* d_in, const int* in_sizes, int n_in,
                              void* d_out, int out_size, void* d_ws, size_t ws_size,
                              hipStream_t stream) {
  const float* coords   = (const float*)d_in[0];
  const float* partners = (const float*)d_in[1];
  const float* props    = (const float*)d_in[2];
  const int*   desc     = (const int*)d_in[3];
  const int*   pairs    = (const int*)d_in[4];
  float*       out      = (float*)d_out;

  const int n_pairs = in_sizes[4] / 2;   // atomPairs is [N_PAIRS,2]
  const int threads = 256;               // 8 wave32 waves per block
  const int blocks  = (n_pairs + threads - 1) / threads;

  electro_pairs_kernel<<<blocks, threads, 0, stream>>>(
      coords, partners, props, desc, pairs, out, n_pairs);
}